// NeatNetworkGPU_33406255628584
// MI455X (gfx1250) — compile-verified
//
#include <hip/hip_runtime.h>
#include <hip/hip_bf16.h>
#include <math.h>

// ---------------------------------------------------------------------------
// NEAT network on MI455X (gfx1250).
//   vals_T: [11264 rows (nodes)] x [1024 cols (batch)], f32, lives in d_ws.
//   Layer kernel: one node per block (blockIdx.y), 8 blocks per node (blockIdx.x), 8 waves x 16-batch tiles.
//   Per-node dot over FANIN=16 done with 4 chained V_WMMA_F32_16X16X4_F32:
//     A[m][k] = w[k] (broadcast rows), B[k][b] = vals_T[src[k]][b]
//     -> D[m][b] = dot(w, gathered[b]) in every row m; lane reads acc[0].
//   Full f32 precision; workload is L2-gather bound (46MB working set < 192MB L2).
// ---------------------------------------------------------------------------

typedef __attribute__((ext_vector_type(2))) float v2f;
typedef __attribute__((ext_vector_type(8))) float v8f;

#define NB      1024      // batch
#define NPLN    2048      // nodes per layer
#define FANIN_N 16

// Generic 32x32-tiled transpose: dst[c*R + r] = src[r*C + c]
__global__ __launch_bounds__(256)
void neat_transpose(const float* __restrict__ src, float* __restrict__ dst,
                    int R, int C)
{
    __shared__ float tile[32][33];
    const int tx = threadIdx.x;          // 0..31
    const int ty = threadIdx.y;          // 0..7
    const int r0 = blockIdx.y * 32;
    const int c0 = blockIdx.x * 32;
#pragma unroll
    for (int i = 0; i < 32; i += 8)
        tile[ty + i][tx] = src[(size_t)(r0 + ty + i) * C + (c0 + tx)];
    __syncthreads();
#pragma unroll
    for (int i = 0; i < 32; i += 8)
        dst[(size_t)(c0 + ty + i) * R + (r0 + tx)] = tile[tx][ty + i];
}

template<int LAST>
__global__ __launch_bounds__(256)
void neat_layer(const float* __restrict__ valsT,   // row-major [rows][1024]
                float*       __restrict__ outRows, // rows for this layer's nodes
                const int*   __restrict__ srcL,    // [2048][16]
                const float* __restrict__ wL,      // [2048][16]
                const float* __restrict__ bL)      // [2048]
{
    const int n     = blockIdx.y;                    // node 0..2047 (block-uniform)
    const int wave  = threadIdx.x >> 5;              // 0..7
    const int lane  = threadIdx.x & 31;
    const int btile = blockIdx.x * 8 + wave;         // 0..63
    const int b0    = btile * 16;
    const int half  = lane >> 4;                     // 0: lanes 0-15, 1: lanes 16-31
    const int col   = b0 + (lane & 15);

    const int*   s  = srcL + n * FANIN_N;            // uniform -> s_load
    const float* ww = wL   + n * FANIN_N;            // uniform -> s_load

    v8f acc = {0.f, 0.f, 0.f, 0.f, 0.f, 0.f, 0.f, 0.f};

#pragma unroll
    for (int kb = 0; kb < FANIN_N; kb += 4) {
        const float w0 = ww[kb + 0], w1 = ww[kb + 1];
        const float w2 = ww[kb + 2], w3 = ww[kb + 3];
        const int   i0 = s[kb + 0],  i1 = s[kb + 1];
        const int   i2 = s[kb + 2],  i3 = s[kb + 3];

        // A (16x4 f32): VGPR0 = K{0|2}, VGPR1 = K{1|3} by lane half; rows broadcast.
        v2f a, b;
        a.x = half ? w2 : w0;
        a.y = half ? w3 : w1;
        // B (4x16 f32): VGPR0 = row K{0|2}, VGPR1 = row K{1|3}; cols = 16 batch.
        const int r0 = half ? i2 : i0;
        const int r1 = half ? i3 : i1;
        b.x = valsT[(size_t)r0 * NB + col];           // coalesced 64B per half-wave
        b.y = valsT[(size_t)r1 * NB + col];

        acc = __builtin_amdgcn_wmma_f32_16x16x4_f32(
            /*neg_a=*/false, a, /*neg_b=*/false, b,
            /*c_mod=*/(short)0, acc, /*reuse_a=*/false, /*reuse_b=*/false);
    }

    // Every lane's acc[0] == dot result for batch (b0 + (lane&15)).
    float r = acc[0] + bL[n];
    r = LAST ? (1.0f / (1.0f + expf(-r))) : tanhf(r);
    if (lane < 16)
        outRows[(size_t)n * NB + col] = r;
}

extern "C" void kernel_launch(void* const* d_in, const int* in_sizes, int n_in,
                              void* d_out, int out_size, void* d_ws, size_t ws_size,
                              hipStream_t stream)
{
    const float* inputs   = (const float*)d_in[0];   // [1024,1024] f32
    const int*   edge_src = (const int*)  d_in[1];   // [5,2048,16] i32
    const float* edge_w   = (const float*)d_in[2];   // [5,2048,16] f32
    const float* biases   = (const float*)d_in[3];   // [5,2048]    f32
    float*       out      = (float*)d_out;           // [1024,256]  f32

    float* valsT = (float*)d_ws;                     // [11264][1024] f32 = 46.1 MB

    // 1) inputs [B=1024][1024] -> vals_T rows 0..1023
    neat_transpose<<<dim3(1024 / 32, 1024 / 32), dim3(32, 8), 0, stream>>>(
        inputs, valsT, /*R=*/1024, /*C=*/1024);

    // 2) five layers, sequential (layer l reads rows < hi, writes rows hi..hi+2047)
    int hi = 1024;
    for (int l = 0; l < 5; ++l) {
        const dim3 grid(NB / (16 * 8), NPLN);        // (8, 2048)
        const dim3 blk(256);
        const int*   sL = edge_src + (size_t)l * NPLN * FANIN_N;
        const float* wL = edge_w   + (size_t)l * NPLN * FANIN_N;
        const float* bL = biases   + (size_t)l * NPLN;
        float* outRows  = valsT + (size_t)hi * NB;
        if (l == 4)
            neat_layer<1><<<grid, blk, 0, stream>>>(valsT, outRows, sL, wL, bL);
        else
            neat_layer<0><<<grid, blk, 0, stream>>>(valsT, outRows, sL, wL, bL);
        hi += NPLN;
    }

    // 3) last 256 rows (11008..11263) -> out[b][j]
    neat_transpose<<<dim3(1024 / 32, 256 / 32), dim3(32, 8), 0, stream>>>(
        valsT + (size_t)(1024 + 5 * NPLN - 256) * NB, out, /*R=*/256, /*C=*/1024);
}